// KruskalLinearLayer_83726092468963
// MI455X (gfx1250) — compile-verified
//
#include <hip/hip_runtime.h>

typedef __attribute__((ext_vector_type(2))) float v2f;
typedef __attribute__((ext_vector_type(8))) float v8f;

#define DIN  4096
#define DOUT 4096

// ---------------------------------------------------------------------------
// Precompute kernel: u_i = f_i @ c_i (six 16x16), then emit Vt and Ut in the
// WMMA-B-operand permuted layout:
//   buf[(g*32 + lane)*2 + slot] = B[k = 4*g + (lane<16 ? 0:2) + slot][n = lane%16]
// so the main kernel's B operand is one contiguous float2 load per lane.
// ---------------------------------------------------------------------------
__global__ __launch_bounds__(256) void kruskal_precompute(
    const float* __restrict__ c0, const float* __restrict__ c1,
    const float* __restrict__ c2, const float* __restrict__ c3,
    const float* __restrict__ c4, const float* __restrict__ c5,
    const float* __restrict__ f0, const float* __restrict__ f1,
    const float* __restrict__ f2, const float* __restrict__ f3,
    const float* __restrict__ f4, const float* __restrict__ f5,
    float* __restrict__ VtP, float* __restrict__ UtP) {
  __shared__ float u[6][16][16];
  const float* cs[6] = {c0, c1, c2, c3, c4, c5};
  const float* fs[6] = {f0, f1, f2, f3, f4, f5};
  const int t = threadIdx.x;       // 256 threads
  const int a = t >> 4, r = t & 15;
#pragma unroll
  for (int i = 0; i < 6; ++i) {
    float s = 0.f;
#pragma unroll
    for (int p = 0; p < 8; ++p) s += fs[i][a * 8 + p] * cs[i][p * 16 + r];
    u[i][a][r] = s;
  }
  __syncthreads();

  // 65536 entries each for VtP and UtP; 256 per thread.
  for (int it = 0; it < 256; ++it) {
    const int idx  = t + (it << 8);
    const int g    = idx >> 6;        // K-group (k0/4) for Vt; (jt*4+rg) for Ut
    const int rem  = idx & 63;
    const int l    = rem >> 1;        // lane 0..31
    const int slot = rem & 1;
    const int kk0  = (l < 16) ? 0 : 2;
    // Vt[k][r] = u3[d][r]*u4[e][r]*u5[f][r], k = d*256+e*16+f
    {
      const int k  = (g << 2) + kk0 + slot;
      const int rr = l & 15;
      const int d = k >> 8, e = (k >> 4) & 15, f = k & 15;
      VtP[idx] = u[3][d][rr] * u[4][e][rr] * u[5][f][rr];
    }
    // Ut[r][j] = u0[a][r]*u1[b][r]*u2[c][r], j = a*256+b*16+c
    {
      const int jt = g >> 2, rg = g & 3;
      const int rr = (rg << 2) + kk0 + slot;
      const int j  = (jt << 4) + (l & 15);
      const int aa = j >> 8, bb = (j >> 4) & 15, cc = j & 15;
      UtP[idx] = u[0][aa][rr] * u[1][bb][rr] * u[2][cc][rr];
    }
  }
}

// ---------------------------------------------------------------------------
// Main kernel: one block per 16-row batch tile, 8 waves (wave32).
// Stage 1: t(16x16) = Xtile(16x4096) @ Vt(4096x16), K split across waves.
// Stage 2: Ytile(16x4096) = t @ Ut(16x4096) + bias, columns split across waves.
// ---------------------------------------------------------------------------
__global__ __launch_bounds__(256) void kruskal_main(
    const float* __restrict__ x, const float* __restrict__ VtP,
    const float* __restrict__ UtP, const float* __restrict__ bias,
    float* __restrict__ out) {
  __shared__ float lds_part[8 * 256];  // per-wave partial 16x16 t tiles
  __shared__ float lds_t[256];         // final t in WMMA-A permuted layout

  const int tid  = threadIdx.x;
  const int lane = tid & 31;
  const int wave = tid >> 5;
  const int m    = lane & 15;                // M (row within tile) / N column
  const int kk0  = (lane < 16) ? 0 : 2;      // K-half for this lane
  const int rowBase = blockIdx.x << 4;       // first batch row of this tile

  // ---- Stage 1: partial t over this wave's K-chunk [wave*512, wave*512+512)
  const float* xrow = x + (size_t)(rowBase + m) * DIN + wave * 512 + kk0;
  const float* vp   = VtP + (size_t)wave * 128 * 64 + (lane << 1);
  v8f acc = {};
#pragma unroll 4
  for (int k0 = 0; k0 < 512; k0 += 4) {
    v2f av = *(const v2f*)(xrow + k0);          // A: X[m][k0+kk0 .. +1]
    v2f bv = *(const v2f*)(vp + (k0 << 4));     // B: Vt rows k0+kk0, +1; col r=m
    acc = __builtin_amdgcn_wmma_f32_16x16x4_f32(
        false, av, false, bv, (short)0, acc, false, false);
  }
  // scatter partial tile: acc[j] -> (M = j + (lane<16?0:8), N = lane%16)
  {
    const int rb = (lane < 16) ? 0 : 8;
#pragma unroll
    for (int j = 0; j < 8; ++j)
      lds_part[wave * 256 + (rb + j) * 16 + m] = acc[j];
  }
  __syncthreads();

  // ---- Reduce 8 partials; write t in WMMA-A permuted layout:
  // lds_t[rg*64 + l*2 + slot] = t[M = l%16][r = rg*4 + (l<16?0:2) + slot]
  {
    float s = 0.f;
#pragma unroll
    for (int w = 0; w < 8; ++w) s += lds_part[w * 256 + tid];
    const int mm = tid >> 4, r = tid & 15;
    const int rg = r >> 2, kk = r & 3;
    const int l  = ((kk >= 2) ? 16 : 0) + mm;
    lds_t[rg * 64 + l * 2 + (kk & 1)] = s;
  }
  __syncthreads();

  // ---- Stage 2: Y = t @ Ut + bias
  v2f ta[4];
#pragma unroll
  for (int rg = 0; rg < 4; ++rg)
    ta[rg] = *(const v2f*)(&lds_t[rg * 64 + (lane << 1)]);

  float* outrow = out + (size_t)rowBase * DOUT;
  for (int jt = wave; jt < 256; jt += 8) {      // 32 column tiles per wave
    v8f c = {};
    const float* up = UtP + (size_t)jt * 4 * 64 + (lane << 1);
#pragma unroll
    for (int rg = 0; rg < 4; ++rg) {
      v2f bv = *(const v2f*)(up + rg * 64);
      c = __builtin_amdgcn_wmma_f32_16x16x4_f32(
          false, ta[rg], false, bv, (short)0, c, false, false);
    }
    const int col = (jt << 4) + m;
    const float bvadd = bias[col];
    const int rb = (lane < 16) ? 0 : 8;
#pragma unroll
    for (int j = 0; j < 8; ++j)
      outrow[(size_t)(rb + j) * DOUT + col] = c[j] + bvadd;
  }
}

// ---------------------------------------------------------------------------
extern "C" void kernel_launch(void* const* d_in, const int* in_sizes, int n_in,
                              void* d_out, int out_size, void* d_ws, size_t ws_size,
                              hipStream_t stream) {
  const float* x = (const float*)d_in[0];
  const float* c0 = (const float*)d_in[1];
  const float* c1 = (const float*)d_in[2];
  const float* c2 = (const float*)d_in[3];
  const float* c3 = (const float*)d_in[4];
  const float* c4 = (const float*)d_in[5];
  const float* c5 = (const float*)d_in[6];
  const float* f0 = (const float*)d_in[7];
  const float* f1 = (const float*)d_in[8];
  const float* f2 = (const float*)d_in[9];
  const float* f3 = (const float*)d_in[10];
  const float* f4 = (const float*)d_in[11];
  const float* f5 = (const float*)d_in[12];
  const float* bias = (const float*)d_in[13];

  float* VtP = (float*)d_ws;        // 4096*16 floats = 256 KB
  float* UtP = VtP + 65536;         // 16*4096 floats = 256 KB

  const int nbatch = in_sizes[0] / DIN;   // 4096
  const int ntiles = nbatch >> 4;         // 256

  kruskal_precompute<<<1, 256, 0, stream>>>(c0, c1, c2, c3, c4, c5,
                                            f0, f1, f2, f3, f4, f5, VtP, UtP);
  kruskal_main<<<ntiles, 256, 0, stream>>>(x, VtP, UtP, bias, (float*)d_out);
}